// Maxlayer_9251359556271
// MI455X (gfx1250) — compile-verified
//
#include <hip/hip_runtime.h>
#include <stdint.h>

// Tropical (max-plus) matmul: out[n,m] = max_k(|x[n,k]| + |w[m,k]|)
// N=1024, K=256, M=1024, fp32. VALU-bound; WMMA does not apply (max-plus
// semiring). CDNA5 path used: async global->LDS copies (ASYNCcnt) with
// double buffering + wave32 register-tiled VALU with |.| source modifiers.

#define N_DIM 1024
#define K_DIM 256
#define M_DIM 1024

constexpr int TILE    = 64;            // output tile is TILE x TILE
constexpr int KT      = 32;            // k-chunk
constexpr int RS      = 36;            // padded LDS row stride (floats): 16B-aligned, conflict-free
constexpr int TILE_F  = TILE * RS;     // 2304 floats per tile
constexpr int STAGE_F = 2 * TILE_F;    // x tile + w tile = 4608 floats per stage
constexpr int NCHUNK  = K_DIM / KT;    // 8

// Async DMA of 16 bytes per lane from global address g to LDS address lds_off, tracked by ASYNCcnt.
__device__ __forceinline__ void async_b128(unsigned lds_off, const float* g) {
  asm volatile("global_load_async_to_lds_b128 %0, %1, off"
               :: "v"(lds_off), "v"(g)
               : "memory");
}

__device__ __forceinline__ float comp4(const float4& v, int k) {
  return (k == 0) ? v.x : (k == 1) ? v.y : (k == 2) ? v.z : v.w;
}

__global__ __launch_bounds__(256)
void tropical_mm_kernel(const float* __restrict__ x,
                        const float* __restrict__ w,
                        float* __restrict__ out) {
  __shared__ __align__(16) float smem[2 * STAGE_F];   // 36 KB, double buffered

  const int tx  = threadIdx.x;            // 0..15 -> M direction
  const int ty  = threadIdx.y;            // 0..15 -> N direction
  const int tid = ty * 16 + tx;
  const int n0  = blockIdx.y * TILE;
  const int m0  = blockIdx.x * TILE;

  // Low 32 bits of the generic pointer to LDS == wave-relative LDS byte offset
  // (flat LDS aperture truncation, ISA 10.2). This is what the async vdst wants.
  const unsigned lds_base = (unsigned)(uintptr_t)(void*)smem;

  // Copy-lane decomposition: 512 16B segments per 64x32 tile -> 2 per thread.
  const int r0 = tid >> 3;     // row 0..31  (second half: +32)
  const int s0 = tid & 7;      // 16B segment within a 128B row

  auto issue_stage = [&](int buf, int k0) {
    const unsigned xbase = lds_base + (unsigned)buf * (STAGE_F * 4);
    const unsigned wbase = xbase + TILE_F * 4;
#pragma unroll
    for (int h = 0; h < 2; ++h) {
      const int r = r0 + 32 * h;
      async_b128(xbase + (unsigned)(r * RS + s0 * 4) * 4u,
                 x + (size_t)(n0 + r) * K_DIM + k0 + s0 * 4);
    }
#pragma unroll
    for (int h = 0; h < 2; ++h) {
      const int r = r0 + 32 * h;
      async_b128(wbase + (unsigned)(r * RS + s0 * 4) * 4u,
                 w + (size_t)(m0 + r) * K_DIM + k0 + s0 * 4);
    }
  };

  float acc[4][4];
#pragma unroll
  for (int i = 0; i < 4; ++i)
#pragma unroll
    for (int j = 0; j < 4; ++j)
      acc[i][j] = 0.0f;                  // |x|+|w| >= 0, so 0 is a safe identity

  issue_stage(0, 0);                     // prologue: stage 0 in flight (4 ops/wave)

  for (int kc = 0; kc < NCHUNK; ++kc) {
    const int buf = kc & 1;

    if (kc + 1 < NCHUNK) {
      issue_stage(buf ^ 1, (kc + 1) * KT);         // prefetch next (4 more ops)
      asm volatile("s_wait_asynccnt 4" ::: "memory");  // current stage landed
    } else {
      asm volatile("s_wait_asynccnt 0" ::: "memory");
    }
    __syncthreads();                               // tiles visible to all waves

    const float* __restrict__ X = smem + buf * STAGE_F;            // [64][36]
    const float* __restrict__ W = smem + buf * STAGE_F + TILE_F;   // [64][36]

#pragma unroll
    for (int k4 = 0; k4 < KT; k4 += 4) {
      float4 xa[4], wb[4];
#pragma unroll
      for (int i = 0; i < 4; ++i)        // rows n0 + ty + 16*i  (broadcast in wave)
        xa[i] = *(const float4*)&X[(ty + 16 * i) * RS + k4];
#pragma unroll
      for (int j = 0; j < 4; ++j)        // rows m0 + tx + 16*j  (conflict-free, padded)
        wb[j] = *(const float4*)&W[(tx + 16 * j) * RS + k4];

#pragma unroll
      for (int kk = 0; kk < 4; ++kk)
#pragma unroll
        for (int i = 0; i < 4; ++i) {
          const float xv = __builtin_fabsf(comp4(xa[i], kk));  // folds into |src| modifier
#pragma unroll
          for (int j = 0; j < 4; ++j) {
            const float s = xv + __builtin_fabsf(comp4(wb[j], kk));
            acc[i][j] = fmaxf(acc[i][j], s);                   // v_max_num_f32
          }
        }
    }
    __syncthreads();                     // buffer may be overwritten next iter
  }

  // Epilogue: out[n][m], m coalesced across tx.
#pragma unroll
  for (int i = 0; i < 4; ++i) {
    const int n = n0 + ty + 16 * i;
#pragma unroll
    for (int j = 0; j < 4; ++j) {
      const int m = m0 + tx + 16 * j;
      out[(size_t)n * M_DIM + m] = acc[i][j];
    }
  }
}

extern "C" void kernel_launch(void* const* d_in, const int* in_sizes, int n_in,
                              void* d_out, int out_size, void* d_ws, size_t ws_size,
                              hipStream_t stream) {
  (void)in_sizes; (void)n_in; (void)out_size; (void)d_ws; (void)ws_size;
  const float* x = (const float*)d_in[0];   // [N, K] fp32
  const float* w = (const float*)d_in[1];   // [M, K] fp32
  float* out     = (float*)d_out;           // [N, M] fp32

  dim3 block(16, 16, 1);
  dim3 grid(M_DIM / TILE, N_DIM / TILE, 1);
  hipLaunchKernelGGL(tropical_mm_kernel, grid, block, 0, stream, x, w, out);
}